// HighPassLoss_17583596110201
// MI455X (gfx1250) — compile-verified
//
#include <hip/hip_runtime.h>
#include <stdint.h>

// ---------------- problem constants (from reference) ----------------
#define Bn     8
#define Hn     256
#define Wn     256
#define TCn    48                  // T*C = 24*2, contiguous innermost
#define WTC    (Wn * TCn)          // 12288 floats per (b,h) row
#define PLANE  ((size_t)Hn * WTC)  // per-batch plane
#define CH     32                  // h-rows per thread (sliding window)
#define NCH    (Hn / CH)           // 8
#define STRIPS (WTC / 4)           // 3072 float4 strips per row
#define BLOCK  256
#define NBLOCKS ((Bn * NCH * STRIPS) / BLOCK)   // 768
#define NTOT   (25165824.0)        // 8*256*256*24*2

typedef float vf4 __attribute__((ext_vector_type(4)));
typedef float v2f __attribute__((ext_vector_type(2)));
typedef float v8f __attribute__((ext_vector_type(8)));

// high-pass partial-sum kernel: each thread slides a float4 strip down CH rows.
__global__ __launch_bounds__(BLOCK) void hp_partial(const float* __restrict__ X,
                                                    const float* __restrict__ Y,
                                                    float* __restrict__ ws) {
    const int tid = blockIdx.x * BLOCK + threadIdx.x;
    const int u4  = tid % STRIPS;         // fastest -> coalesced
    const int rr  = tid / STRIPS;         // uniform within a block (3072%256==0)
    const int hc  = rr % NCH;
    const int b   = rr / NCH;
    const int u   = u4 * 4;               // byte offset 16*u4 -> 16B aligned
    const int w   = u / TCn;
    const int h0  = hc * CH;

    // branchless w-edge handling: clamp the neighbor address (harmless reload
    // of the center cacheline) and kill its contribution with a 0/1 mask fma.
    const bool lOK = (w > 0);
    const bool rOK = (w < Wn - 1);
    const int   uoffL = lOK ? -TCn : 0;
    const int   uoffR = rOK ?  TCn : 0;
    const float lm = lOK ? 1.f : 0.f;
    const float rm = rOK ? 1.f : 0.f;

    const float* __restrict__ xb = X + (size_t)b * PLANE + u;
    const float* __restrict__ yb = Y + (size_t)b * PLANE + u;

    // unmasked row load: 6 back-to-back global_load_b128, no branches.
    auto loadRowFast = [&](int h, vf4& rs, vf4& dc) {
        const float* px = xb + (size_t)h * WTC;
        const float* py = yb + (size_t)h * WTC;
        vf4 xc = *(const vf4*)px;
        vf4 yc = *(const vf4*)py;
        vf4 xm = *(const vf4*)(px + uoffL);
        vf4 ym = *(const vf4*)(py + uoffL);
        vf4 xp = *(const vf4*)(px + uoffR);
        vf4 yp = *(const vf4*)(py + uoffR);
        dc = xc - yc;
        vf4 dm = xm - ym;
        vf4 dp = xp - yp;
        rs = dm * lm + dc;    // contracts to v_fma (mask fold, same cost as add)
        rs = dp * rm + rs;
    };

    // masked version: only used for rows -1 and 256 (block-uniform branch).
    auto loadRowSafe = [&](int h, vf4& rs, vf4& dc) {
        if (h < 0 || h >= Hn) {
            vf4 z = {0.f, 0.f, 0.f, 0.f};
            rs = z; dc = z;
        } else {
            loadRowFast(h, rs, dc);
        }
    };

    vf4 r_m, r_c, r_p, d_c, d_p, tmp;
    loadRowSafe(h0 - 1, r_m, tmp);        // top halo (zeros when h0==0)
    loadRowFast(h0,     r_c, d_c);        // h0 in [0,224]: always valid

    float acc = 0.f;
    // hot loop: rows h0 .. h0+CH-2 -> next row h+1 <= h0+CH-1 <= 255 (valid)
    #pragma unroll 4
    for (int h = h0; h < h0 + CH - 1; ++h) {
        int hpf = h + 4;                  // branchless stream-ahead prefetch
        hpf = hpf < Hn ? hpf : Hn - 1;
        __builtin_prefetch(xb + (size_t)hpf * WTC, 0, 1);
        __builtin_prefetch(yb + (size_t)hpf * WTC, 0, 1);

        loadRowFast(h + 1, r_p, d_p);
        vf4 box = r_m + r_c + r_p;        // vertical 3-box of row sums
        vf4 out = 9.f * d_c - box;        // 8*center - 8 neighbors
        acc += __builtin_fabsf(out.x) + __builtin_fabsf(out.y)
             + __builtin_fabsf(out.z) + __builtin_fabsf(out.w);
        r_m = r_c; r_c = r_p; d_c = d_p;
    }
    {   // peeled last row: bottom halo may be out of range (zeros)
        loadRowSafe(h0 + CH, r_p, d_p);
        vf4 box = r_m + r_c + r_p;
        vf4 out = 9.f * d_c - box;
        acc += __builtin_fabsf(out.x) + __builtin_fabsf(out.y)
             + __builtin_fabsf(out.z) + __builtin_fabsf(out.w);
    }

    // ---- wave32 reduction via V_WMMA_F32_16X16X4_F32 ----
    // A(16x4): VGPR0 lanes0-15 -> A[m][0]=acc[m]; lanes16-31 -> A[m][2]=acc[m+16]
    // VGPR1 (K=1/K=3) = 0.  B = all-ones (layout invariant).
    // => D[m][n] = acc[m] + acc[m+16] for every n.
    v2f amat; amat.x = acc; amat.y = 0.f;
    v2f bmat; bmat.x = 1.f; bmat.y = 1.f;
    v8f cz = {0.f, 0.f, 0.f, 0.f, 0.f, 0.f, 0.f, 0.f};
    v8f d = __builtin_amdgcn_wmma_f32_16x16x4_f32(
        /*neg_a=*/false, amat, /*neg_b=*/false, bmat,
        /*c_mod=*/(short)0, cz, /*reuse_a=*/false, /*reuse_b=*/false);
    // lanes 0-15 hold sum over m=0..7, lanes 16-31 hold sum over m=8..15
    float cs = d[0] + d[1] + d[2] + d[3] + d[4] + d[5] + d[6] + d[7];
    union { float f; int i; } ui, uo;
    ui.f = cs;
    uo.i = __builtin_amdgcn_ds_swizzle(ui.i, 0x401F);  // SWAPX16 (group-of-32)
    float wsum = cs + uo.f;                            // full 32-lane sum

    __shared__ float wpart[BLOCK / 32];
    if ((threadIdx.x & 31) == 0) wpart[threadIdx.x >> 5] = wsum;
    __syncthreads();
    if (threadIdx.x == 0) {
        float s = 0.f;
        #pragma unroll
        for (int i = 0; i < BLOCK / 32; ++i) s += wpart[i];
        ws[blockIdx.x] = s;   // deterministic per-block partial
    }
}

// deterministic fixed-order final reduction of NBLOCKS partials
__global__ __launch_bounds__(BLOCK) void hp_final(const float* __restrict__ ws,
                                                  float* __restrict__ out) {
    __shared__ double sh[BLOCK];
    double s = 0.0;
    for (int i = threadIdx.x; i < NBLOCKS; i += BLOCK) s += (double)ws[i];
    sh[threadIdx.x] = s;
    __syncthreads();
    for (int off = BLOCK / 2; off > 0; off >>= 1) {
        if (threadIdx.x < off) sh[threadIdx.x] += sh[threadIdx.x + off];
        __syncthreads();
    }
    if (threadIdx.x == 0) out[0] = (float)(sh[0] * (1.0 / NTOT));
}

extern "C" void kernel_launch(void* const* d_in, const int* in_sizes, int n_in,
                              void* d_out, int out_size, void* d_ws, size_t ws_size,
                              hipStream_t stream) {
    const float* X = (const float*)d_in[0];
    const float* Y = (const float*)d_in[1];
    float* out = (float*)d_out;
    float* ws  = (float*)d_ws;     // needs NBLOCKS*4 = 3 KB scratch

    hp_partial<<<NBLOCKS, BLOCK, 0, stream>>>(X, Y, ws);
    hp_final<<<1, BLOCK, 0, stream>>>(ws, out);
}